// ComplexSKPDELayer_49598282334506
// MI455X (gfx1250) — compile-verified
//
#include <hip/hip_runtime.h>
#include <hip/hip_bf16.h>
#include <math.h>

// ---------------------------------------------------------------------------
// Helmholtz PINN residual evaluator for MI455X (gfx1250), fp32 WMMA path.
//   d_in: coords_int, coords_bc, gt_bc, latent, W1(67x128), b1, W2(128x128),
//         b2, W3(128x2), b3   (all float32, row-major)
//   d_out: [loss_pde, loss_bc, helm_res interleaved re/im x N_INT]
// ---------------------------------------------------------------------------

#define N_INT 262144
#define N_BC  65536
#define LAT   64
#define HID   128
#define K2    4.0f   // K=2 -> K^2 = 4 (real)

typedef __attribute__((ext_vector_type(2))) float v2f;
typedef __attribute__((ext_vector_type(8))) float v8f;

// fp32 WMMA: D(16x16) = A(16x4) * B(4x16) + C   (wave32)
__device__ __forceinline__ v8f wmma4(v2f a, v2f b, v8f c) {
  return __builtin_amdgcn_wmma_f32_16x16x4_f32(
      /*neg_a=*/false, a, /*neg_b=*/false, b,
      /*c_mod=*/(short)0, c, /*reuse_a=*/false, /*reuse_b=*/false);
}

// LDS layout constants (float offsets)
#define W2_STRIDE 144                    // 128 + 16 pad: halves hit disjoint banks
#define T_STRIDE  132                    // 128 + 4 pad: 16 rows -> 16 distinct bank pairs
#define W2S_FLOATS (128 * W2_STRIDE)
#define TILE_FLOATS (16 * T_STRIDE)

// ---------------------------------------------------------------------------
// c1[j] = b1[j] + sum_k latent[k] * W1[3+k][j]   (constant part of layer 1)
// ---------------------------------------------------------------------------
__global__ void c1_kernel(const float* __restrict__ latent,
                          const float* __restrict__ W1,
                          const float* __restrict__ b1,
                          float* __restrict__ c1out) {
  int j = threadIdx.x;  // 128 threads
  float s = b1[j];
  for (int k = 0; k < LAT; ++k) s += latent[k] * W1[(3 + k) * HID + j];
  c1out[j] = s;
}

// ---------------------------------------------------------------------------
// Interior kernel: forward + Laplacian (trace of Hessian) per point.
// Block = 256 threads = 8 waves; each wave handles a 16-point tile.
// N-tiles processed in pairs so each set of 7 A fragments feeds 14 WMMAs.
// ---------------------------------------------------------------------------
__global__ void __launch_bounds__(256)
pde_interior_kernel(const float* __restrict__ coords,
                    const float* __restrict__ W1,
                    const float* __restrict__ W2,
                    const float* __restrict__ b2,
                    const float* __restrict__ W3,
                    const float* __restrict__ b3,
                    const float* __restrict__ c1,
                    float* __restrict__ helm_out,        // interleaved re,im
                    float* __restrict__ block_partials) {
  extern __shared__ float lds[];
  float* W2s  = lds;                       // 128 x 144
  float* W1r  = W2s + W2S_FLOATS;          // 3 x 128
  float* c1s  = W1r + 3 * HID;             // 128
  float* b2s  = c1s + HID;                 // 128
  float* W3s  = b2s + HID;                 // 128 x 2
  float* tilT = W3s + 2 * HID;             // 8 waves x (16 x 132) t = tanh(u1)
  float* tilS = tilT + 8 * TILE_FLOATS;    // 8 waves x (16 x 132) s = 1 - t^2
  float* wsum = tilS + 8 * TILE_FLOATS;    // 8 per-wave loss partials

  const int tid  = threadIdx.x;
  const int wave = tid >> 5;
  const int lane = tid & 31;
  const int lm   = lane & 15;   // matrix column / point row within half
  const int grp  = lane >> 4;   // 0 = lanes 0-15, 1 = lanes 16-31

  // ---- stage weights into LDS (block cooperative) ----
  for (int i = tid; i < HID * HID; i += 256)
    W2s[(i >> 7) * W2_STRIDE + (i & 127)] = W2[i];
  for (int i = tid; i < 3 * HID; i += 256) W1r[i] = W1[i];
  if (tid < HID) { c1s[tid] = c1[tid]; b2s[tid] = b2[tid]; }
  W3s[tid] = W3[tid & 255];  // 256 entries, one per thread
  __syncthreads();

  float* tT = tilT + wave * TILE_FLOATS;
  float* tS = tilS + wave * TILE_FLOATS;

  const int tile = blockIdx.x * 8 + wave;
  const int gm0  = tile * 16;

  // ---- layer 1: u1 = x @ W1[0:3,:] + c1 ; t = tanh(u1), s = 1 - t^2 ----
  for (int e = lane; e < 16 * HID; e += 32) {
    int m = e >> 7, j = e & 127;
    const float* cp = coords + (size_t)(gm0 + m) * 3;
    float u = c1s[j] + cp[0] * W1r[j] + cp[1] * W1r[HID + j] + cp[2] * W1r[2 * HID + j];
    float t = tanhf(u);
    tT[m * T_STRIDE + j] = t;
    tS[m * T_STRIDE + j] = 1.0f - t * t;
  }
  __syncthreads();

  // per-point partial sums over hidden column j (per C-row-slot v)
  float pr[8], pi[8], lr[8], li[8];
#pragma unroll
  for (int v = 0; v < 8; ++v) { pr[v] = 0.f; pi[v] = 0.f; lr[v] = 0.f; li[v] = 0.f; }

  for (int np = 0; np < 4; ++np) {   // pair of N-tiles: cols colA and colA+16
    const int colA = np * 32 + lm;
    const int colB = colA + 16;
    const float b2jA = b2s[colA], b2jB = b2s[colB];
    const float w30A = W3s[colA * 2 + 0], w31A = W3s[colA * 2 + 1];
    const float w30B = W3s[colB * 2 + 0], w31B = W3s[colB * 2 + 1];

    v8f aV0 = {}, aD00 = {}, aD10 = {}, aD20 = {}, aS00 = {}, aS10 = {}, aS20 = {};
    v8f aV1 = {}, aD01 = {}, aD11 = {}, aD21 = {}, aS01 = {}, aS11 = {}, aS21 = {};

    for (int kk = 0; kk < 32; ++kk) {
      const int k0 = kk * 4 + grp * 2;  // K pair handled by this half-wave
      // B fragments for both N-tiles: W2[k0..k0+1][colA], [colB]
      v2f BA, BB;
      BA.x = W2s[k0 * W2_STRIDE + colA];
      BA.y = W2s[(k0 + 1) * W2_STRIDE + colA];
      BB.x = W2s[k0 * W2_STRIDE + colB];
      BB.y = W2s[(k0 + 1) * W2_STRIDE + colB];
      // layer-1 channel data for row lm at K = k0, k0+1 (shared by both tiles)
      const float t0 = tT[lm * T_STRIDE + k0];
      const float t1 = tT[lm * T_STRIDE + k0 + 1];
      const float s0 = tS[lm * T_STRIDE + k0];
      const float s1 = tS[lm * T_STRIDE + k0 + 1];
      const float w00 = W1r[k0],           w01 = W1r[k0 + 1];
      const float w10 = W1r[HID + k0],     w11 = W1r[HID + k0 + 1];
      const float w20 = W1r[2 * HID + k0], w21 = W1r[2 * HID + k0 + 1];
      const float q0 = -2.0f * t0 * s0, q1 = -2.0f * t1 * s1;

      v2f Av;  Av.x  = t0;             Av.y  = t1;
      v2f Ad0; Ad0.x = s0 * w00;       Ad0.y = s1 * w01;
      v2f Ad1; Ad1.x = s0 * w10;       Ad1.y = s1 * w11;
      v2f Ad2; Ad2.x = s0 * w20;       Ad2.y = s1 * w21;
      v2f As0; As0.x = q0 * w00 * w00; As0.y = q1 * w01 * w01;
      v2f As1; As1.x = q0 * w10 * w10; As1.y = q1 * w11 * w11;
      v2f As2; As2.x = q0 * w20 * w20; As2.y = q1 * w21 * w21;

      aV0  = wmma4(Av,  BA, aV0);   aV1  = wmma4(Av,  BB, aV1);
      aD00 = wmma4(Ad0, BA, aD00);  aD01 = wmma4(Ad0, BB, aD01);
      aD10 = wmma4(Ad1, BA, aD10);  aD11 = wmma4(Ad1, BB, aD11);
      aD20 = wmma4(Ad2, BA, aD20);  aD21 = wmma4(Ad2, BB, aD21);
      aS00 = wmma4(As0, BA, aS00);  aS01 = wmma4(As0, BB, aS01);
      aS10 = wmma4(As1, BA, aS10);  aS11 = wmma4(As1, BB, aS11);
      aS20 = wmma4(As2, BA, aS20);  aS21 = wmma4(As2, BB, aS21);
    }

    // layer-2 activation transform + layer-3 (128 -> 2) partial accumulation
#pragma unroll
    for (int v = 0; v < 8; ++v) {
      {
        float a2 = tanhf(aV0[v] + b2jA);
        float s2 = 1.0f - a2 * a2;
        float lap = s2 * (aS00[v] + aS10[v] + aS20[v]) -
                    2.0f * a2 * s2 *
                        (aD00[v] * aD00[v] + aD10[v] * aD10[v] + aD20[v] * aD20[v]);
        pr[v] += a2 * w30A;  pi[v] += a2 * w31A;
        lr[v] += lap * w30A; li[v] += lap * w31A;
      }
      {
        float a2 = tanhf(aV1[v] + b2jB);
        float s2 = 1.0f - a2 * a2;
        float lap = s2 * (aS01[v] + aS11[v] + aS21[v]) -
                    2.0f * a2 * s2 *
                        (aD01[v] * aD01[v] + aD11[v] * aD11[v] + aD21[v] * aD21[v]);
        pr[v] += a2 * w30B;  pi[v] += a2 * w31B;
        lr[v] += lap * w30B; li[v] += lap * w31B;
      }
    }
  }

  // reduce across the 16 lanes of each half-wave (xor masks < 16 stay in-half)
#pragma unroll
  for (int v = 0; v < 8; ++v) {
    for (int off = 8; off >= 1; off >>= 1) {
      pr[v] += __shfl_xor(pr[v], off, 32);
      pi[v] += __shfl_xor(pi[v], off, 32);
      lr[v] += __shfl_xor(lr[v], off, 32);
      li[v] += __shfl_xor(li[v], off, 32);
    }
  }

  float local = 0.0f;
  if (lm == 0) {  // lanes 0 (rows 0..7) and 16 (rows 8..15)
    const float b30 = b3[0], b31 = b3[1];
#pragma unroll
    for (int v = 0; v < 8; ++v) {
      int gm = gm0 + v + grp * 8;
      float p0  = pr[v] + b30;
      float p1  = pi[v] + b31;
      float hre = lr[v] + K2 * p0;
      float him = li[v] + K2 * p1;
      helm_out[2 * (size_t)gm]     = hre;
      helm_out[2 * (size_t)gm + 1] = him;
      local += hre * hre + him * him;
    }
  }
  local += __shfl_xor(local, 16, 32);     // combine the two half-wave owners
  if (lane == 0) wsum[wave] = local;
  __syncthreads();
  if (tid == 0) {
    float s = 0.f;
#pragma unroll
    for (int w = 0; w < 8; ++w) s += wsum[w];  // fixed order -> deterministic
    block_partials[blockIdx.x] = s;
  }
}

// ---------------------------------------------------------------------------
// Boundary kernel: forward only. 4 N-tiles per K-sweep (1 A load -> 4 WMMAs).
// ---------------------------------------------------------------------------
__global__ void __launch_bounds__(256)
bc_kernel(const float* __restrict__ coords,
          const float* __restrict__ gt,
          const float* __restrict__ W1,
          const float* __restrict__ W2,
          const float* __restrict__ b2,
          const float* __restrict__ W3,
          const float* __restrict__ b3,
          const float* __restrict__ c1,
          float* __restrict__ block_partials) {
  extern __shared__ float lds[];
  float* W2s  = lds;                       // 128 x 144
  float* W1r  = W2s + W2S_FLOATS;          // 3 x 128
  float* c1s  = W1r + 3 * HID;             // 128
  float* b2s  = c1s + HID;                 // 128
  float* W3s  = b2s + HID;                 // 256
  float* tilT = W3s + 2 * HID;             // 8 x (16 x 132)
  float* wsum = tilT + 8 * TILE_FLOATS;    // 8

  const int tid  = threadIdx.x;
  const int wave = tid >> 5;
  const int lane = tid & 31;
  const int lm   = lane & 15;
  const int grp  = lane >> 4;

  for (int i = tid; i < HID * HID; i += 256)
    W2s[(i >> 7) * W2_STRIDE + (i & 127)] = W2[i];
  for (int i = tid; i < 3 * HID; i += 256) W1r[i] = W1[i];
  if (tid < HID) { c1s[tid] = c1[tid]; b2s[tid] = b2[tid]; }
  W3s[tid] = W3[tid & 255];
  __syncthreads();

  float* tT = tilT + wave * TILE_FLOATS;
  const int tile = blockIdx.x * 8 + wave;
  const int gm0  = tile * 16;

  for (int e = lane; e < 16 * HID; e += 32) {
    int m = e >> 7, j = e & 127;
    const float* cp = coords + (size_t)(gm0 + m) * 3;
    float u = c1s[j] + cp[0] * W1r[j] + cp[1] * W1r[HID + j] + cp[2] * W1r[2 * HID + j];
    tT[m * T_STRIDE + j] = tanhf(u);
  }
  __syncthreads();

  float pr[8], pi[8];
#pragma unroll
  for (int v = 0; v < 8; ++v) { pr[v] = 0.f; pi[v] = 0.f; }

  for (int np = 0; np < 2; ++np) {   // 4 N-tiles per pass
    const int c0 = np * 64 + lm;
    v8f ac0 = {}, ac1 = {}, ac2 = {}, ac3 = {};

    for (int kk = 0; kk < 32; ++kk) {
      const int k0 = kk * 4 + grp * 2;
      v2f A;
      A.x = tT[lm * T_STRIDE + k0];
      A.y = tT[lm * T_STRIDE + k0 + 1];
      v2f B0, B1, B2v, B3;
      B0.x  = W2s[k0 * W2_STRIDE + c0];        B0.y  = W2s[(k0 + 1) * W2_STRIDE + c0];
      B1.x  = W2s[k0 * W2_STRIDE + c0 + 16];   B1.y  = W2s[(k0 + 1) * W2_STRIDE + c0 + 16];
      B2v.x = W2s[k0 * W2_STRIDE + c0 + 32];   B2v.y = W2s[(k0 + 1) * W2_STRIDE + c0 + 32];
      B3.x  = W2s[k0 * W2_STRIDE + c0 + 48];   B3.y  = W2s[(k0 + 1) * W2_STRIDE + c0 + 48];
      ac0 = wmma4(A, B0, ac0);
      ac1 = wmma4(A, B1, ac1);
      ac2 = wmma4(A, B2v, ac2);
      ac3 = wmma4(A, B3, ac3);
    }

#define BC_EPI(ACC, COL)                                   \
    {                                                      \
      const int col_ = (COL);                              \
      const float b2j_ = b2s[col_];                        \
      const float w30_ = W3s[col_ * 2 + 0];                \
      const float w31_ = W3s[col_ * 2 + 1];                \
      _Pragma("unroll")                                    \
      for (int v = 0; v < 8; ++v) {                        \
        float a2_ = tanhf((ACC)[v] + b2j_);                \
        pr[v] += a2_ * w30_;                               \
        pi[v] += a2_ * w31_;                               \
      }                                                    \
    }
    BC_EPI(ac0, c0)
    BC_EPI(ac1, c0 + 16)
    BC_EPI(ac2, c0 + 32)
    BC_EPI(ac3, c0 + 48)
#undef BC_EPI
  }

#pragma unroll
  for (int v = 0; v < 8; ++v) {
    for (int off = 8; off >= 1; off >>= 1) {
      pr[v] += __shfl_xor(pr[v], off, 32);
      pi[v] += __shfl_xor(pi[v], off, 32);
    }
  }

  float local = 0.0f;
  if (lm == 0) {
    const float b30 = b3[0], b31 = b3[1];
#pragma unroll
    for (int v = 0; v < 8; ++v) {
      int gm = gm0 + v + grp * 8;
      float dr = (pr[v] + b30) - gt[gm];
      float di = pi[v] + b31;
      local += dr * dr + di * di;
    }
  }
  local += __shfl_xor(local, 16, 32);
  if (lane == 0) wsum[wave] = local;
  __syncthreads();
  if (tid == 0) {
    float s = 0.f;
#pragma unroll
    for (int w = 0; w < 8; ++w) s += wsum[w];
    block_partials[blockIdx.x] = s;
  }
}

// ---------------------------------------------------------------------------
// Final deterministic reduction of block partials -> losses.
// ---------------------------------------------------------------------------
__global__ void __launch_bounds__(256)
finalize_kernel(const float* __restrict__ partials, float* __restrict__ out) {
  __shared__ float sh[256];
  int tid = threadIdx.x;

  float s = 0.f;
  for (int i = tid; i < 2048; i += 256) s += partials[i];
  sh[tid] = s;
  __syncthreads();
  for (int o = 128; o > 0; o >>= 1) {
    if (tid < o) sh[tid] += sh[tid + o];
    __syncthreads();
  }
  if (tid == 0) out[0] = sh[0] / (float)N_INT;
  __syncthreads();

  s = 0.f;
  for (int i = tid; i < 512; i += 256) s += partials[2048 + i];
  sh[tid] = s;
  __syncthreads();
  for (int o = 128; o > 0; o >>= 1) {
    if (tid < o) sh[tid] += sh[tid + o];
    __syncthreads();
  }
  if (tid == 0) out[1] = sh[0] / (float)N_BC;
}

// ---------------------------------------------------------------------------
extern "C" void kernel_launch(void* const* d_in, const int* in_sizes, int n_in,
                              void* d_out, int out_size, void* d_ws, size_t ws_size,
                              hipStream_t stream) {
  (void)in_sizes; (void)n_in; (void)out_size; (void)ws_size;
  const float* coords_int = (const float*)d_in[0];
  const float* coords_bc  = (const float*)d_in[1];
  const float* gt_bc      = (const float*)d_in[2];
  const float* latent     = (const float*)d_in[3];
  const float* W1         = (const float*)d_in[4];
  const float* b1         = (const float*)d_in[5];
  const float* W2         = (const float*)d_in[6];
  const float* b2         = (const float*)d_in[7];
  const float* W3         = (const float*)d_in[8];
  const float* b3         = (const float*)d_in[9];

  float* out = (float*)d_out;
  float* ws  = (float*)d_ws;
  float* part_int = ws;          // 2048 floats
  float* part_bc  = ws + 2048;   // 512 floats
  float* c1       = ws + 2560;   // 128 floats

  c1_kernel<<<1, 128, 0, stream>>>(latent, W1, b1, c1);

  const size_t lds_int =
      (size_t)(W2S_FLOATS + 3 * HID + HID + HID + 2 * HID + 16 * TILE_FLOATS + 8) * 4;
  pde_interior_kernel<<<N_INT / 128, 256, lds_int, stream>>>(
      coords_int, W1, W2, b2, W3, b3, c1, out + 2, part_int);

  const size_t lds_bc =
      (size_t)(W2S_FLOATS + 3 * HID + HID + HID + 2 * HID + 8 * TILE_FLOATS + 8) * 4;
  bc_kernel<<<N_BC / 128, 256, lds_bc, stream>>>(
      coords_bc, gt_bc, W1, W2, b2, W3, b3, c1, part_bc);

  finalize_kernel<<<1, 256, 0, stream>>>(ws, out);
}